// ResizeFlow_71554155151982
// MI455X (gfx1250) — compile-verified
//
#include <hip/hip_runtime.h>

// ResizeFlow: (2,3,96,96,96) f32 -> (2,3,192,192,192) f32
// trilinear, factor 2, circular (dft) wrap, then *2 displacement rescale.
// Per-axis weights collapse to {0.25,0.75} by output parity.

#define ISZ 96
#define OSZ 192
#define TILE_OY 16
#define KY_PER_TILE 8                    // TILE_OY / 2
#define LDS_ROWS 10                      // KY_PER_TILE + 2 (wrap halo)
#define LDS_PLANE (LDS_ROWS * ISZ)       // 960 floats per z-plane
#define LDS_TOTAL (2 * LDS_PLANE)        // 1920 floats = 7.5 KB
#define CHUNKS_PER_ROW (ISZ / 4)         // 24 x b128 (16B) chunks per row
#define CHUNKS_TOTAL (2 * LDS_ROWS * CHUNKS_PER_ROW) // 480

typedef float v2f __attribute__((ext_vector_type(2)));
typedef float v4f __attribute__((ext_vector_type(4)));

__global__ __launch_bounds__(256)
void ResizeFlow_kernel(const float* __restrict__ in, float* __restrict__ out)
{
    __shared__ float tile[LDS_TOTAL];
    const int tid = threadIdx.x;

    // block -> (bc, oz, oy-tile)
    unsigned t = blockIdx.x;
    const int oyt = (int)(t % (OSZ / TILE_OY)); t /= (OSZ / TILE_OY);
    const int oz  = (int)(t % OSZ);             t /= OSZ;
    const int bc  = (int)t;                     // 0..5 (batch*channel)

    // z-axis: two source planes + weights from parity
    const int   kz  = oz >> 1, pz = oz & 1;
    const int   izA = pz ? kz : (kz + ISZ - 1) % ISZ;
    const int   izB = pz ? (kz + 1) % ISZ : kz;
    const float wz0 = pz ? 0.75f : 0.25f;
    const float wz1 = 1.0f - wz0;

    const int ky0 = oyt * KY_PER_TILE;       // first ky of this tile
    const int oy0 = oyt * TILE_OY;

    // ---------- async global -> LDS stage of 2 x 10 x 96 input tile ----------
    const unsigned ldsbase = (unsigned)(unsigned long long)(&tile[0]);
    for (int e = tid; e < CHUNKS_TOTAL; e += 256) {
        const int zsel = e / (LDS_ROWS * CHUNKS_PER_ROW);
        const int rem  = e % (LDS_ROWS * CHUNKS_PER_ROW);
        const int yy   = rem / CHUNKS_PER_ROW;      // 0..9 -> input y = ky0-1+yy (wrapped)
        const int xc   = rem % CHUNKS_PER_ROW;      // 16-byte chunk within row
        const int iy   = (ky0 + yy + ISZ - 1) % ISZ;
        const int iz   = zsel ? izB : izA;
        const unsigned goff =
            ((((unsigned)bc * ISZ + (unsigned)iz) * ISZ + (unsigned)iy) * ISZ
             + (unsigned)xc * 4u) * 4u;             // byte offset
        const unsigned loff = ldsbase + (unsigned)e * 16u;
        asm volatile("global_load_async_to_lds_b128 %0, %1, %2"
                     :
                     : "v"(loff), "v"(goff), "s"(in)
                     : "memory");
    }
    asm volatile("s_wait_asynccnt 0" ::: "memory");
    __syncthreads();

    // ---------- compute 16 x 192 output tile: 3 float4 stores / thread ----------
    const unsigned obase = (((unsigned)bc * OSZ + (unsigned)oz) * OSZ
                            + (unsigned)oy0) * OSZ;          // divisible by 4
    v4f* __restrict__ out4 = (v4f*)out + (obase >> 2);

    #pragma unroll
    for (int j = 0; j < 3; ++j) {
        const int q   = j * 256 + tid;       // float4 index in tile: 0..767
        const int oyl = q / (OSZ / 4);       // /48 -> output row 0..15
        const int m   = q % (OSZ / 4);       // float4 column group; ox = 4m..4m+3

        const int   oy  = oy0 + oyl;
        const int   ky  = oy >> 1, py = oy & 1;
        const int   rA  = ky - ky0 + py;     // LDS row (halo at row 0)
        const int   rB  = rA + 1;
        const float wyA = py ? 0.75f : 0.25f;
        const float wyB = 1.0f - wyA;

        // 4 consecutive outputs touch input x in {2m-1, 2m, 2m+1, 2m+2}
        const int xm = (2 * m + ISZ - 1) % ISZ;   // 2m-1 wrapped
        const int xc = 2 * m;                     // even -> 8B-aligned pair load
        const int xp = (2 * m + 2) % ISZ;         // 2m+2 wrapped

        float a0 = 0.f, a1 = 0.f, a2 = 0.f, a3 = 0.f;
        #pragma unroll
        for (int p = 0; p < 2; ++p) {
            const float wz = p ? wz1 : wz0;
            const int   pl = p ? LDS_PLANE : 0;
            #pragma unroll
            for (int r = 0; r < 2; ++r) {
                const float w  = wz * (r ? wyB : wyA);
                const int   ro = pl + (r ? rB : rA) * ISZ;     // even
                const float Lm = tile[ro + xm];
                const v2f   Lc = *(const v2f*)&tile[ro + xc];  // {L0, L1}
                const float Lp = tile[ro + xp];
                a0 += w * (0.25f * Lm   + 0.75f * Lc.x);
                a1 += w * (0.75f * Lc.x + 0.25f * Lc.y);
                a2 += w * (0.25f * Lc.x + 0.75f * Lc.y);
                a3 += w * (0.75f * Lc.y + 0.25f * Lp);
            }
        }
        v4f v;
        v.x = 2.0f * a0; v.y = 2.0f * a1; v.z = 2.0f * a2; v.w = 2.0f * a3;
        __builtin_nontemporal_store(v, out4 + q);   // streaming write-once output
    }
}

extern "C" void kernel_launch(void* const* d_in, const int* in_sizes, int n_in,
                              void* d_out, int out_size, void* d_ws, size_t ws_size,
                              hipStream_t stream)
{
    (void)n_in; (void)out_size; (void)d_ws; (void)ws_size;
    const float* in  = (const float*)d_in[0];
    float*       out = (float*)d_out;
    const int bc = in_sizes[0] / (ISZ * ISZ * ISZ);           // batch*channels = 6
    const unsigned blocks = (unsigned)bc * OSZ * (OSZ / TILE_OY);
    ResizeFlow_kernel<<<dim3(blocks), dim3(256), 0, stream>>>(in, out);
}